// Predictor_57389353009381
// MI455X (gfx1250) — compile-verified
//
#include <hip/hip_runtime.h>
#include <hip/hip_bf16.h>
#include <math.h>

#define N_NODES 100000
#define E_EDGES 1600000
#define F_IN    118
#define CH      64
#define HEADS   8
#define CDIM    8
#define LAYERS  4
#define GBATCH  512
#define NEG_SLOPE 0.2f
#define ETOT (E_EDGES + N_NODES)   // 1,700,000 edges incl. self loops

typedef __attribute__((ext_vector_type(2))) float v2f;
typedef __attribute__((ext_vector_type(8))) float v8f;

// ---- monotone float<->uint key for atomic segment-max of signed floats ----
__device__ __forceinline__ unsigned fkey(float f) {
    unsigned u = __float_as_uint(f);
    return (u & 0x80000000u) ? ~u : (u | 0x80000000u);
}
__device__ __forceinline__ float funkey(unsigned k) {
    return (k & 0x80000000u) ? __uint_as_float(k ^ 0x80000000u)
                             : __uint_as_float(~k);
}

// =====================================================================
// Dense input layer: h = x @ W[118,64] + b.  One block per node; the
// 118-float x row is staged in LDS, each thread produces one channel.
// =====================================================================
__global__ __launch_bounds__(64)
void dense_kernel(const float* __restrict__ x, const float* __restrict__ W,
                  const float* __restrict__ b, float* __restrict__ h) {
    __shared__ float xs[F_IN];
    const int n = blockIdx.x;
    const int t = threadIdx.x;
    for (int k = t; k < F_IN; k += 64) xs[k] = x[n * F_IN + k];
    __syncthreads();
    float acc = b[t];
    #pragma unroll 2
    for (int k = 0; k < F_IN; ++k) acc += xs[k] * W[k * CH + t];
    h[n * CH + t] = acc;
}

// =====================================================================
// Layer transform GEMM: out[N,64] = A[N,64] @ W[64,64] + bias, fp32 WMMA.
// One wave per 16x16 output tile; 16 steps of V_WMMA_F32_16X16X4_F32.
// A layout (16x4 f32): lanes 0-15 -> K = k,k+1 ; lanes 16-31 -> K = k+2,k+3.
// B layout (4x16 f32): mirrored; C/D: VGPR r = rows r (lanes 0-15) / r+8.
// N = 100000 = 6250*16 exactly -> no predication, EXEC all ones.
// =====================================================================
__global__ __launch_bounds__(256)
void gemm64_wmma(const float* __restrict__ A, const float* __restrict__ W,
                 const float* __restrict__ bias, float* __restrict__ out) {
    const int wave  = threadIdx.x >> 5;
    const int lane  = threadIdx.x & 31;
    const int tile  = blockIdx.x * 8 + wave;   // (N/16)*4 tiles total
    const int rt    = tile >> 2;               // row tile
    const int ct    = tile & 3;                // col tile (of 4)
    const int row   = lane & 15;
    const int khalf = lane >> 4;               // 0 or 1

    v8f acc = {};
    const float* arow = A + (size_t)(rt * 16 + row) * CH;
    const int    col  = ct * 16 + row;

    #pragma unroll
    for (int k = 0; k < CH; k += 4) {
        const int kk = k + khalf * 2;
        v2f a;
        a.x = arow[kk];
        a.y = arow[kk + 1];
        v2f bm;
        bm.x = W[kk * CH + col];
        bm.y = W[(kk + 1) * CH + col];
        acc = __builtin_amdgcn_wmma_f32_16x16x4_f32(
            /*neg_a=*/false, a, /*neg_b=*/false, bm,
            /*c_mod=*/(short)0, acc, /*reuse_a=*/false, /*reuse_b=*/false);
    }

    const float bb = bias[col];
    #pragma unroll
    for (int r = 0; r < 8; ++r) {
        const int m = r + khalf * 8;
        out[(size_t)(rt * 16 + m) * CH + col] = acc[r] + bb;
    }
}

// =====================================================================
// Per-layer init: out <- conv_bias broadcast (aggregation accumulator),
// smax keys <- 0 (below key(-inf)), denom <- 0.
// =====================================================================
__global__ __launch_bounds__(256)
void layer_init(float* __restrict__ out, const float* __restrict__ cbias,
                unsigned* __restrict__ smax, float* __restrict__ denom) {
    const int i = blockIdx.x * blockDim.x + threadIdx.x;
    if (i < N_NODES * CH) out[i] = cbias[i & (CH - 1)];
    if (i < N_NODES * HEADS) { smax[i] = 0u; denom[i] = 0.f; }
}

// =====================================================================
// Edge pass 1: score[e,h] = att_h . leakyrelu(xl[src] + xr[dst]);
// atomic segment-max into smax keys. One thread per (edge, head).
// =====================================================================
__global__ __launch_bounds__(256)
void edge_score(const int* __restrict__ ei, const float* __restrict__ xl,
                const float* __restrict__ xr, const float* __restrict__ attl,
                float* __restrict__ score, unsigned* __restrict__ smax) {
    const int i = blockIdx.x * blockDim.x + threadIdx.x;
    if (i >= ETOT * HEADS) return;
    const int e = i >> 3;
    const int hh = i & 7;
    int s, d;
    if (e < E_EDGES) { s = ei[e]; d = ei[E_EDGES + e]; }
    else             { s = d = e - E_EDGES; }
    const float* pl = xl + (size_t)s * CH + hh * CDIM;
    const float* pr = xr + (size_t)d * CH + hh * CDIM;
    float sc = 0.f;
    #pragma unroll
    for (int c = 0; c < CDIM; ++c) {
        float v = pl[c] + pr[c];
        v = (v > 0.f) ? v : v * NEG_SLOPE;
        sc += attl[hh * CDIM + c] * v;
    }
    score[i] = sc;
    atomicMax(&smax[d * HEADS + hh], fkey(sc));
}

// =====================================================================
// Edge pass 2: ex = exp(score - smax[dst]); overwrite score; atomic
// segment-sum into denom.
// =====================================================================
__global__ __launch_bounds__(256)
void edge_exp(const int* __restrict__ ei, float* __restrict__ score,
              const unsigned* __restrict__ smax, float* __restrict__ denom) {
    const int i = blockIdx.x * blockDim.x + threadIdx.x;
    if (i >= ETOT * HEADS) return;
    const int e = i >> 3;
    const int hh = i & 7;
    int d;
    if (e < E_EDGES) d = ei[E_EDGES + e]; else d = e - E_EDGES;
    const float m  = funkey(smax[d * HEADS + hh]);
    const float ex = __expf(score[i] - m);
    score[i] = ex;
    atomicAdd(&denom[d * HEADS + hh], ex);
}

// =====================================================================
// Edge pass 3: alpha = ex / denom[dst]; out[dst] += alpha * xl[src].
// out was pre-initialized to conv_bias, so out is the next h directly.
// =====================================================================
__global__ __launch_bounds__(256)
void edge_agg(const int* __restrict__ ei, const float* __restrict__ xl,
              const float* __restrict__ score, const float* __restrict__ denom,
              float* __restrict__ out) {
    const int i = blockIdx.x * blockDim.x + threadIdx.x;
    if (i >= ETOT * HEADS) return;
    const int e = i >> 3;
    const int hh = i & 7;
    int s, d;
    if (e < E_EDGES) { s = ei[e]; d = ei[E_EDGES + e]; }
    else             { s = d = e - E_EDGES; }
    const float alpha = score[i] / (denom[d * HEADS + hh] + 1e-16f);
    const float* pl = xl + (size_t)s * CH + hh * CDIM;
    float* po = out + (size_t)d * CH + hh * CDIM;
    #pragma unroll
    for (int c = 0; c < CDIM; ++c) atomicAdd(&po[c], alpha * pl[c]);
}

// =====================================================================
// Pooling + head
// =====================================================================
__global__ __launch_bounds__(256)
void pool_init(float* __restrict__ pooled, float* __restrict__ cnt) {
    const int i = blockIdx.x * blockDim.x + threadIdx.x;
    if (i < GBATCH * CH) pooled[i] = 0.f;
    if (i < GBATCH) cnt[i] = 0.f;
}

__global__ __launch_bounds__(256)
void pool_accum(const float* __restrict__ h, const int* __restrict__ batch,
                float* __restrict__ pooled, float* __restrict__ cnt) {
    const int i = blockIdx.x * blockDim.x + threadIdx.x;
    if (i >= N_NODES * CH) return;
    const int n = i >> 6;
    const int c = i & (CH - 1);
    const int b = batch[n];
    atomicAdd(&pooled[b * CH + c], h[i]);
    if (c == 0) atomicAdd(&cnt[b], 1.f);
}

__global__ __launch_bounds__(256)
void head_kernel(const float* __restrict__ pooled, const float* __restrict__ cnt,
                 const float* __restrict__ hW, const float* __restrict__ hb,
                 float* __restrict__ out) {
    const int g = blockIdx.x * blockDim.x + threadIdx.x;
    if (g >= GBATCH) return;
    const float inv = 1.f / fmaxf(cnt[g], 1.f);
    float acc = hb[0];
    #pragma unroll
    for (int c = 0; c < CH; ++c) acc += pooled[g * CH + c] * inv * hW[c];
    out[g] = sinhf(acc);
}

// =====================================================================
extern "C" void kernel_launch(void* const* d_in, const int* in_sizes, int n_in,
                              void* d_out, int out_size, void* d_ws, size_t ws_size,
                              hipStream_t stream) {
    const float* x     = (const float*)d_in[0];
    const int*   ei    = (const int*)d_in[1];
    const int*   batch = (const int*)d_in[2];
    const float* dW    = (const float*)d_in[3];
    const float* db    = (const float*)d_in[4];
    const float* Wl    = (const float*)d_in[5];
    const float* bl    = (const float*)d_in[6];
    const float* Wr    = (const float*)d_in[7];
    const float* br    = (const float*)d_in[8];
    const float* att   = (const float*)d_in[9];
    const float* cb    = (const float*)d_in[10];
    const float* hW    = (const float*)d_in[11];
    const float* hb    = (const float*)d_in[12];
    float* out = (float*)d_out;

    // workspace carve (floats)
    float* ws = (float*)d_ws;
    size_t off = 0;
    float* hA     = ws + off; off += (size_t)N_NODES * CH;     // 6.4M
    float* hB     = ws + off; off += (size_t)N_NODES * CH;     // 6.4M
    float* xlbuf  = ws + off; off += (size_t)N_NODES * CH;     // 6.4M
    float* xrbuf  = ws + off; off += (size_t)N_NODES * CH;     // 6.4M
    float* score  = ws + off; off += (size_t)ETOT * HEADS;     // 13.6M
    unsigned* smax = (unsigned*)(ws + off); off += (size_t)N_NODES * HEADS;
    float* denom  = ws + off; off += (size_t)N_NODES * HEADS;
    float* pooled = ws + off; off += (size_t)GBATCH * CH;
    float* cnt    = ws + off; off += (size_t)GBATCH;

    // 1) dense input layer
    dense_kernel<<<N_NODES, 64, 0, stream>>>(x, dW, db, hA);

    const int gemmBlocks = (N_NODES / 16) * 4 / 8;           // 3125, exact
    const int eThreads   = ETOT * HEADS;                     // 13.6M
    const int eBlocks    = (eThreads + 255) / 256;
    const int nBlocks    = (N_NODES * CH + 255) / 256;

    float* hcur = hA;
    float* hnext = hB;
    for (int l = 0; l < LAYERS; ++l) {
        gemm64_wmma<<<gemmBlocks, 256, 0, stream>>>(hcur, Wl + l * CH * CH,
                                                    bl + l * CH, xlbuf);
        gemm64_wmma<<<gemmBlocks, 256, 0, stream>>>(hcur, Wr + l * CH * CH,
                                                    br + l * CH, xrbuf);
        layer_init<<<nBlocks, 256, 0, stream>>>(hnext, cb + l * CH, smax, denom);
        edge_score<<<eBlocks, 256, 0, stream>>>(ei, xlbuf, xrbuf,
                                                att + l * HEADS * CDIM, score, smax);
        edge_exp<<<eBlocks, 256, 0, stream>>>(ei, score, smax, denom);
        edge_agg<<<eBlocks, 256, 0, stream>>>(ei, xlbuf, score, denom, hnext);
        float* t = hcur; hcur = hnext; hnext = t;
    }

    pool_init<<<(GBATCH * CH + 255) / 256, 256, 0, stream>>>(pooled, cnt);
    pool_accum<<<nBlocks, 256, 0, stream>>>(hcur, batch, pooled, cnt);
    head_kernel<<<(GBATCH + 255) / 256, 256, 0, stream>>>(pooled, cnt, hW, hb, out);
}